// TVConv_test_47914655154337
// MI455X (gfx1250) — compile-verified
//
#include <hip/hip_runtime.h>

// Problem constants (match reference setup_inputs()).
#define BATCH 8
#define CH    96
#define H     128
#define W     128
#define ROWS  4                    // output rows per block
#define TROWS (ROWS + 2)           // 6 rows incl. row halo
#define COL0  4                    // interior (gw=0) starts at this LDS column -> 16B aligned
#define TSTRIDE 136                // floats per LDS row (544 B), keeps b128 chunks aligned
#define TILE_ELEMS (TROWS * TSTRIDE)
#define THREADS 128
#define CHUNKS (TROWS * 32)        // 32 x 16B chunks per 512B row = 192 chunks

// ---- CDNA5 async global->LDS copies (gfx1250), ASYNCcnt-tracked ----
__device__ __forceinline__ void async_load_b128_to_lds(unsigned int lds_byte_off,
                                                       const float* gaddr) {
    asm volatile("global_load_async_to_lds_b128 %0, %1, off"
                 :
                 : "v"(lds_byte_off),
                   "v"((unsigned long long)(uintptr_t)gaddr)
                 : "memory");
}

__device__ __forceinline__ void wait_async_zero() {
    asm volatile("s_wait_asynccnt 0" ::: "memory");
}

__global__ __launch_bounds__(THREADS)
void tvconv_gfx1250_kernel(const float* __restrict__ x,
                           const float* __restrict__ wm,
                           float* __restrict__ out) {
    // Double-buffered x strip in LDS (2 * 6 * 136 * 4B = 6528 B).
    __shared__ float tile[2][TILE_ELEMS];

    const int tid = threadIdx.x;            // == column w (0..127)
    const int w   = tid;
    const int tilesPerC = H / ROWS;         // 32
    const int c   = blockIdx.x / tilesPerC;
    const int h0  = (blockIdx.x % tilesPerC) * ROWS;

    // Zero both buffers once (float4). Halo cells (cols COL0-1 and COL0+W, and
    // rows whose gh is outside the image) are never overwritten by async loads,
    // so this implements the conv's zero padding for free.
    {
        float4* t4 = (float4*)&tile[0][0];
        const float4 z = make_float4(0.f, 0.f, 0.f, 0.f);
        for (int e = tid; e < (2 * TILE_ELEMS) / 4; e += THREADS) t4[e] = z;
    }

    // Per-pixel weights -> registers, loaded ONCE, reused for all 8 batches.
    // weight_maps layout (1,C,3,3,H,W): idx = ((c*9 + ij)*H + h)*W + w (coalesced in w)
    float wr[ROWS][9];
#pragma unroll
    for (int r = 0; r < ROWS; ++r) {
#pragma unroll
        for (int ij = 0; ij < 9; ++ij) {
            wr[r][ij] = wm[((c * 9 + ij) * H + (h0 + r)) * W + w];
        }
    }

    __syncthreads();   // zeros in place before any async writes land

    // Stage the 6-row strip of batch b into LDS buffer `buf` with b128 async
    // copies: 32 chunks of 16B per row, 192 chunks total -> <=2 ops/thread.
    auto issue_loads = [&](int b, int buf) {
        const float* xb = x + (size_t)(b * CH + c) * (H * W);
#pragma unroll
        for (int pass = 0; pass < 2; ++pass) {
            const int chunk = tid + pass * THREADS;
            if (chunk < CHUNKS) {
                const int row = chunk >> 5;       // 0..5
                const int k   = chunk & 31;       // 16B chunk within the row
                const int gh  = h0 - 1 + row;
                if (gh >= 0 && gh < H) {
                    unsigned int lds_off =
                        (unsigned int)(uintptr_t)&tile[buf][row * TSTRIDE + COL0 + 4 * k];
                    async_load_b128_to_lds(lds_off, xb + gh * W + 4 * k);
                }
            }
        }
    };

    // Prologue: fill buffer 0 with batch 0.
    issue_loads(0, 0);
    wait_async_zero();
    __syncthreads();

    for (int b = 0; b < BATCH; ++b) {
        const int cur = b & 1;

        // Prefetch next batch into the other buffer (overlaps compute).
        if (b + 1 < BATCH) issue_loads(b + 1, cur ^ 1);

        const float* t = &tile[cur][0];
#pragma unroll
        for (int r = 0; r < ROWS; ++r) {
            float acc = 0.0f;
#pragma unroll
            for (int i = 0; i < 3; ++i) {
                // output col w reads gw = w-1..w+1  ->  LDS cols (w+COL0-1)..(w+COL0+1)
                const float* trow = t + (r + i) * TSTRIDE + (w + COL0 - 1);
                acc = fmaf(wr[r][i * 3 + 0], trow[0], acc);
                acc = fmaf(wr[r][i * 3 + 1], trow[1], acc);
                acc = fmaf(wr[r][i * 3 + 2], trow[2], acc);
            }
            out[((size_t)(b * CH + c) * H + (h0 + r)) * W + w] = acc;
        }

        if (b + 1 < BATCH) {
            wait_async_zero();   // our prefetch done
            __syncthreads();     // everyone's prefetch done; safe to swap buffers
        }
    }
}

extern "C" void kernel_launch(void* const* d_in, const int* in_sizes, int n_in,
                              void* d_out, int out_size, void* d_ws, size_t ws_size,
                              hipStream_t stream) {
    const float* x  = (const float*)d_in[0];   // (8,96,128,128) f32
    const float* wm = (const float*)d_in[1];   // (1,96,3,3,128,128) f32
    float* out      = (float*)d_out;           // (8,96,128,128) f32

    const int grid = CH * (H / ROWS);          // 96 * 32 = 3072 blocks
    tvconv_gfx1250_kernel<<<grid, THREADS, 0, stream>>>(x, wm, out);
}